// KaldiFbank_1726576857718
// MI455X (gfx1250) — compile-verified
//
#include <hip/hip_runtime.h>
#include <math.h>

typedef __attribute__((ext_vector_type(2))) float v2f;
typedef __attribute__((ext_vector_type(4))) float v4f;
typedef __attribute__((ext_vector_type(8))) float v8f;

#define WINLEN   400
#define SHIFT    160
#define NFFT     512
#define NBIN     256      // bins 0..255 (bin 256 has zero mel weight in reference)
#define NMEL     80
#define MFRM     998      // 1 + (160000-400)/160
#define MGROUPS  32       // 32 frames per block -> 32*32 = 1024 >= 998
#define SAMPLES  160000
#define PREEMPH  0.97f
#define EPSF     1.1920929e-07f
#define PI2F     6.28318530717958647692f

// ---- workspace layout (in floats) -------------------------------------------
// povey window   : 512             @ OFF_POV
// cos+sin table  : 128*16*32*4     @ OFF_CS  (fragment-packed, interleaved:
//                  [kstep][ntile][lane]{c_r0, c_r1, s_r0, s_r1} -> b128/lane)
// mel weights    : 64*5*32*2       @ OFF_W   (fragment-packed: [kstep][jtile][lane][2])
#define OFF_POV 0
#define OFF_CS  512
#define OFF_W   (512 + 262144)

static __device__ __forceinline__ v8f wmma_f32x4(v2f a, v2f b, v8f c) {
  return __builtin_amdgcn_wmma_f32_16x16x4_f32(false, a, false, b, (short)0, c,
                                               false, false);
}

// ---- init: DFT cos/sin table, interleaved fragment-packed -------------------
// B-fragment mapping for V_WMMA 16x16x4 f32: reg r, lane l -> K = 2*(l>>4)+r,
// N = l&15 within the 4x16 tile.
__global__ void k_init_dft(float* __restrict__ CS) {
  int t = blockIdx.x * blockDim.x + threadIdx.x;       // 0 .. 65535
  if (t >= 128 * 16 * 32) return;
  int lane = t & 31;
  int nt   = (t >> 5) & 15;
  int ks   = t >> 9;                                   // 0..127
  int n    = nt * 16 + (lane & 15);
  int hi   = lane >> 4;
#pragma unroll
  for (int r = 0; r < 2; ++r) {
    int k    = ks * 4 + 2 * hi + r;                    // 0..511
    int ph   = (k * n) & (NFFT - 1);                   // exact mod-512 reduction
    float a  = -PI2F * (float)ph * (1.0f / (float)NFFT);
    CS[4 * t + r]     = cosf(a);
    CS[4 * t + 2 + r] = sinf(a);
  }
}

// ---- init: mel weights, fragment-packed -------------------------------------
__global__ void k_init_mel(float* __restrict__ W) {
  int t = blockIdx.x * blockDim.x + threadIdx.x;       // 0 .. 10239
  if (t >= 64 * 5 * 32) return;
  int lane = t & 31;
  int jt   = (t >> 5) % 5;
  int ks   = t / 160;                                  // 0..63
  int j    = jt * 16 + (lane & 15);                    // 0..79
  int hi   = lane >> 4;
  const float ml = 1127.0f * logf(1.0f + 20.0f / 700.0f);
  const float mh = 1127.0f * logf(1.0f + 8000.0f / 700.0f);
  const float dd = (mh - ml) / (float)(NMEL + 1);
#pragma unroll
  for (int r = 0; r < 2; ++r) {
    int kb    = ks * 4 + 2 * hi + r;                   // FFT bin 0..255
    float fr  = (float)kb * (16000.0f / 512.0f);
    float mm  = 1127.0f * logf(1.0f + fr / 700.0f);
    float lft = ml + (float)j * dd;
    float up  = (mm - lft) / dd;
    float dn  = (lft + 2.0f * dd - mm) / dd;
    W[2 * t + r] = fmaxf(0.0f, fminf(up, dn));
  }
}

// ---- init: Povey window -----------------------------------------------------
__global__ void k_init_pov(float* __restrict__ pov) {
  int t = blockIdx.x * blockDim.x + threadIdx.x;
  if (t >= 512) return;
  float v = 0.0f;
  if (t < WINLEN) {
    float h = 0.5f - 0.5f * cosf(PI2F * (float)t / (float)(WINLEN - 1));
    v = powf(h, 0.85f);
  }
  pov[t] = v;
}

// ---- fused fbank kernel -----------------------------------------------------
// Block: 32 frames (two 16-row M tiles) of one batch element.  8 waves.
//   Phase 0: mean / preemph / window -> LDS Xt[32][516]
//   Phase 1: wave w owns n-tiles {2w,2w+1}; one b128 fetch per (ntile) per
//            K-step delivers cos+sin fragments; 8 f32 WMMAs per 2 fetches.
//            power -> LDS Pt[32][260]
//   Phase 2: 10 mel tile-tasks (5 j-tiles x 2 M-halves) over 8 waves,
//            K=256 WMMA loop, log(max(.,eps)), transposed store.
__global__ __launch_bounds__(256) void k_fbank(
    const float* __restrict__ wav, const float* __restrict__ pov,
    const float* __restrict__ CSp, const float* __restrict__ Wp,
    float* __restrict__ out) {
  __shared__ float Xt[32 * 516];
  __shared__ float Pt[32 * 260];

  const int mg   = blockIdx.x;       // M group (32 frames)
  const int b    = blockIdx.y;       // batch
  const int tid  = threadIdx.x;
  const int lane = tid & 31;
  const int wid  = tid >> 5;

  // ---------------- Phase 0: windowing into LDS ----------------
  {
    int f    = tid >> 3;             // frame within group, 0..31
    int i    = tid & 7;              // lane within 8-thread frame team
    int gm   = mg * 32 + f;
    int base = (gm < MFRM ? gm : MFRM - 1) * SHIFT;   // clamp: junk rows discarded later
    const float* x = wav + (size_t)b * SAMPLES + base;

    float s = 0.0f;
#pragma unroll
    for (int q = 0; q < 50; ++q) s += x[i + 8 * q];   // 400 = 8*50
#pragma unroll
    for (int o = 4; o; o >>= 1) s += __shfl_xor(s, o, 8);
    float mean = s * (1.0f / (float)WINLEN);

    for (int k = i; k < NFFT; k += 8) {
      float v = 0.0f;
      if (k < WINLEN) {
        float xk = x[k];
        float xp = (k > 0) ? x[k - 1] : x[0];
        // (d[k] - p*d[k-1]) = xk - p*xp - (1-p)*mean
        v = (xk - PREEMPH * xp - (1.0f - PREEMPH) * mean) * pov[k];
      }
      Xt[f * 516 + k] = v;
    }
  }
  __syncthreads();

  // ---------------- Phase 1: DFT (cos+sin) + power ----------------
  {
    const int nt0 = wid * 2, nt1 = nt0 + 1;
    const int m   = lane & 15;
    const int hi  = lane >> 4;
    v8f c0a = {}, s0a = {}, c1a = {}, s1a = {};
    v8f c0b = {}, s0b = {}, c1b = {}, s1b = {};
    const v4f* CS4 = (const v4f*)CSp;
    for (int ks = 0; ks < 128; ++ks) {
      int xo = ks * 4 + 2 * hi;
      v2f aA  = *(const v2f*)&Xt[m * 516 + xo];
      v2f aB  = *(const v2f*)&Xt[(16 + m) * 516 + xo];
      v4f f0  = CS4[(ks * 16 + nt0) * 32 + lane];   // one b128: cos+sin frag
      v4f f1  = CS4[(ks * 16 + nt1) * 32 + lane];
      v2f bc0 = __builtin_shufflevector(f0, f0, 0, 1);
      v2f bs0 = __builtin_shufflevector(f0, f0, 2, 3);
      v2f bc1 = __builtin_shufflevector(f1, f1, 0, 1);
      v2f bs1 = __builtin_shufflevector(f1, f1, 2, 3);
      c0a = wmma_f32x4(aA, bc0, c0a);
      s0a = wmma_f32x4(aA, bs0, s0a);
      c1a = wmma_f32x4(aA, bc1, c1a);
      s1a = wmma_f32x4(aA, bs1, s1a);
      c0b = wmma_f32x4(aB, bc0, c0b);
      s0b = wmma_f32x4(aB, bs0, s0b);
      c1b = wmma_f32x4(aB, bc1, c1b);
      s1b = wmma_f32x4(aB, bs1, s1b);
    }
    // C/D layout: reg r, lane l -> M = r + 8*(l>>4), N = l&15
    int pm = hi * 8;
    int n0 = nt0 * 16 + m;
    int n1 = nt1 * 16 + m;
#pragma unroll
    for (int r = 0; r < 8; ++r) {
      Pt[(pm + r) * 260 + n0]        = c0a[r] * c0a[r] + s0a[r] * s0a[r];
      Pt[(pm + r) * 260 + n1]        = c1a[r] * c1a[r] + s1a[r] * s1a[r];
      Pt[(16 + pm + r) * 260 + n0]   = c0b[r] * c0b[r] + s0b[r] * s0b[r];
      Pt[(16 + pm + r) * 260 + n1]   = c1b[r] * c1b[r] + s1b[r] * s1b[r];
    }
  }
  __syncthreads();

  // ---------------- Phase 2: mel + log + transposed store ----------------
  {
    const int m  = lane & 15;
    const int hi = lane >> 4;
    const v2f* W2 = (const v2f*)Wp;
    for (int task = wid; task < 10; task += 8) {   // wave-uniform loop
      int jt   = task % 5;
      int half = task / 5;
      v8f acc = {};
      for (int ks = 0; ks < 64; ++ks) {
        v2f a  = *(const v2f*)&Pt[(half * 16 + m) * 260 + ks * 4 + 2 * hi];
        v2f bw = W2[(ks * 5 + jt) * 32 + lane];
        acc = wmma_f32x4(a, bw, acc);
      }
      int j     = jt * 16 + m;                      // mel index 0..79
      int mbase = mg * 32 + half * 16 + hi * 8;
#pragma unroll
      for (int r = 0; r < 8; ++r) {
        int gm = mbase + r;
        if (gm < MFRM)
          out[((size_t)b * NMEL + j) * MFRM + gm] = logf(fmaxf(acc[r], EPSF));
      }
    }
  }
}

// ---- launcher ---------------------------------------------------------------
extern "C" void kernel_launch(void* const* d_in, const int* in_sizes, int n_in,
                              void* d_out, int out_size, void* d_ws,
                              size_t ws_size, hipStream_t stream) {
  (void)in_sizes; (void)n_in; (void)out_size; (void)ws_size;
  const float* wav = (const float*)d_in[0];
  float* out = (float*)d_out;
  float* w   = (float*)d_ws;

  float* pov = w + OFF_POV;
  float* CS  = w + OFF_CS;
  float* W   = w + OFF_W;

  k_init_dft<<<256, 256, 0, stream>>>(CS);
  k_init_mel<<<40, 256, 0, stream>>>(W);
  k_init_pov<<<2, 256, 0, stream>>>(pov);
  k_fbank<<<dim3(MGROUPS, 32), 256, 0, stream>>>(wav, pov, CS, W, out);
}